// GraphLSTMCell_4337916969719
// MI455X (gfx1250) — compile-verified
//
#include <hip/hip_runtime.h>
#include <hip/hip_bf16.h>
#include <math.h>

typedef __attribute__((ext_vector_type(2))) float v2f;
typedef __attribute__((ext_vector_type(8))) float v8f;

#define BB 4
#define NN 325
#define NPAD 336   // 21*16
#define II 64
#define HH 128
#define G4 512     // 4*H
#define KL 192     // I+H
#define MT 21      // 16-row tiles per batch

// workspace offsets (floats)
#define OFF_WG      0u        // [192][512] packed [W_ih^T ; W_hh^T]
#define OFF_WST     98304u    // [128][256] packed [Ws^T | Wt^T]
#define OFF_WCOMB   131072u   // [256][128] comb_w^T
#define OFF_HLSTM   163840u   // [B][NPAD][H]
#define OFF_SUPPORT 335872u   // [B][NPAD][H]
#define OFF_HGRAPH  507904u   // [B][NPAD][H]
#define OFF_ST      679936u   // [B][NPAD][256]  (s | t)
#define OFF_HATT    1024000u  // [B][NPAD][H]
// total ~1.20M floats (~4.8 MB)

__device__ __forceinline__ float sigmoidf_(float x) { return 1.0f / (1.0f + expf(-x)); }

__device__ __forceinline__ v8f wmma4(v2f a, v2f b, v8f c) {
  // V_WMMA_F32_16X16X4_F32: D = A(16x4) * B(4x16) + C(16x16), all fp32
  return __builtin_amdgcn_wmma_f32_16x16x4_f32(false, a, false, b, (short)0, c, false, false);
}

// ---------------- K0: pack weights K-major ----------------
__global__ __launch_bounds__(256) void k_pack(const float* __restrict__ W_ih,
                                              const float* __restrict__ W_hh,
                                              const float* __restrict__ Ws_w,
                                              const float* __restrict__ Wt_w,
                                              const float* __restrict__ comb_w,
                                              float* __restrict__ ws) {
  int idx = blockIdx.x * 256 + threadIdx.x;
  if (idx < KL * G4) {                      // Wg[k][g] = Wih/Whh[g][k]
    int k = idx >> 9, g = idx & 511;
    ws[OFF_WG + idx] = (k < II) ? W_ih[g * II + k] : W_hh[g * HH + (k - II)];
  }
  if (idx < HH * 256) {                     // Wst[k][o] = Ws_w[o][k] / Wt_w[o-128][k]
    int k = idx >> 8, o = idx & 255;
    ws[OFF_WST + idx] = (o < HH) ? Ws_w[o * HH + k] : Wt_w[(o - HH) * HH + k];
  }
  if (idx < 256 * HH) {                     // Wc[k][o] = comb_w[o][k]
    int k = idx >> 7, o = idx & 127;
    ws[OFF_WCOMB + idx] = comb_w[o * 256 + k];
  }
}

// ---------------- K1: fused LSTM cell (gates GEMM + nonlinearity) ----------------
__global__ __launch_bounds__(256) void k_lstm(const float* __restrict__ x,
                                              const float* __restrict__ h,
                                              const float* __restrict__ c,
                                              const float* __restrict__ b_ih,
                                              const float* __restrict__ b_hh,
                                              float* __restrict__ ws,
                                              float* __restrict__ c_out) {
  __shared__ float As[16][KL];   // 12 KB A strip [x|h]
  __shared__ float Gs[16][G4];   // 32 KB gates strip
  const float* Wg = ws + OFF_WG;
  float* hlstm = ws + OFF_HLSTM;
  int b = blockIdx.x / MT, m0 = (blockIdx.x % MT) * 16;
  for (int t = threadIdx.x; t < 16 * KL; t += 256) {
    int r = t / KL, kk = t - r * KL;
    int row = m0 + r;
    float vv = 0.f;
    if (row < NN) vv = (kk < II) ? x[(b * NN + row) * II + kk]
                                 : h[(b * NN + row) * HH + (kk - II)];
    As[r][kk] = vv;
  }
  __syncthreads();
  int lane = threadIdx.x & 31, wave = threadIdx.x >> 5;
  int mn = lane & 15, kh = (lane >> 4) << 1;
  int col0 = wave * 64;                      // each wave: 4 column tiles
  v8f acc[4] = {};
  for (int k = 0; k < KL; k += 4) {
    v2f a; a.x = As[mn][k + kh]; a.y = As[mn][k + kh + 1];
#pragma unroll
    for (int t4 = 0; t4 < 4; ++t4) {
      int cc = col0 + t4 * 16 + mn;
      v2f bf; bf.x = Wg[(k + kh) * G4 + cc]; bf.y = Wg[(k + kh + 1) * G4 + cc];
      acc[t4] = wmma4(a, bf, acc[t4]);
    }
  }
#pragma unroll
  for (int t4 = 0; t4 < 4; ++t4) {
    int cc = col0 + t4 * 16 + mn;
#pragma unroll
    for (int r = 0; r < 8; ++r) Gs[(lane >> 4) * 8 + r][cc] = acc[t4][r];
  }
  __syncthreads();
  for (int t = threadIdx.x; t < 16 * HH; t += 256) {
    int r = t >> 7, hh = t & 127;
    int row = m0 + r;
    float ig = Gs[r][hh]          + b_ih[hh]          + b_hh[hh];
    float fg = Gs[r][HH + hh]     + b_ih[HH + hh]     + b_hh[HH + hh];
    float gg = Gs[r][2 * HH + hh] + b_ih[2 * HH + hh] + b_hh[2 * HH + hh];
    float og = Gs[r][3 * HH + hh] + b_ih[3 * HH + hh] + b_hh[3 * HH + hh];
    float hn = 0.f;
    if (row < NN) {
      float cp = c[(b * NN + row) * HH + hh];
      float cn = sigmoidf_(fg) * cp + sigmoidf_(ig) * tanhf(gg);
      hn = sigmoidf_(og) * tanhf(cn);
      c_out[(b * NN + row) * HH + hh] = cn;
    }
    hlstm[(b * NPAD + row) * HH + hh] = hn;   // pad rows zeroed
  }
}

// ---------------- K2a: support = h_lstm @ gc_w ----------------
__global__ __launch_bounds__(256) void k_support(const float* __restrict__ gc_w,
                                                 float* __restrict__ ws) {
  __shared__ float As[16][HH];
  int b = blockIdx.x / MT, m0 = (blockIdx.x % MT) * 16;
  const float* hlstm = ws + OFF_HLSTM;
  float* sup = ws + OFF_SUPPORT;
  for (int t = threadIdx.x; t < 16 * HH; t += 256) {
    int r = t >> 7, kk = t & 127;
    As[r][kk] = hlstm[(b * NPAD + m0 + r) * HH + kk];
  }
  __syncthreads();
  int lane = threadIdx.x & 31, wave = threadIdx.x >> 5;
  int mn = lane & 15, kh = (lane >> 4) << 1;
  int col = wave * 16 + mn;
  v8f acc = {};
  for (int k = 0; k < HH; k += 4) {
    v2f a; a.x = As[mn][k + kh]; a.y = As[mn][k + kh + 1];
    v2f bf; bf.x = gc_w[(k + kh) * HH + col]; bf.y = gc_w[(k + kh + 1) * HH + col];
    acc = wmma4(a, bf, acc);
  }
#pragma unroll
  for (int r = 0; r < 8; ++r)
    sup[(b * NPAD + m0 + (lane >> 4) * 8 + r) * HH + col] = acc[r];
}

// ---------------- K2b: h_graph = adj @ support + gc_b ----------------
__global__ __launch_bounds__(256) void k_graph(const float* __restrict__ adj,
                                               const float* __restrict__ gc_b,
                                               float* __restrict__ ws) {
  __shared__ float As[16][NPAD];   // 21.5 KB adj strip (zero padded)
  int b = blockIdx.x / MT, m0 = (blockIdx.x % MT) * 16;
  const float* sup = ws + OFF_SUPPORT;
  float* hg = ws + OFF_HGRAPH;
  for (int t = threadIdx.x; t < 16 * NPAD; t += 256) {
    int r = t / NPAD, kk = t - r * NPAD;
    int row = m0 + r;
    As[r][kk] = (row < NN && kk < NN) ? adj[(b * NN + row) * NN + kk] : 0.f;
  }
  __syncthreads();
  int lane = threadIdx.x & 31, wave = threadIdx.x >> 5;
  int mn = lane & 15, kh = (lane >> 4) << 1;
  int col = wave * 16 + mn;
  v8f acc = {};
  for (int k = 0; k < NPAD; k += 4) {
    v2f a; a.x = As[mn][k + kh]; a.y = As[mn][k + kh + 1];
    v2f bf; bf.x = sup[(b * NPAD + k + kh) * HH + col];
    bf.y = sup[(b * NPAD + k + kh + 1) * HH + col];
    acc = wmma4(a, bf, acc);
  }
  float bias = gc_b[col];
#pragma unroll
  for (int r = 0; r < 8; ++r) {
    int row = m0 + (lane >> 4) * 8 + r;
    hg[(b * NPAD + row) * HH + col] = (row < NN) ? (acc[r] + bias) : 0.f;
  }
}

// ---------------- K3: [s|t] = h_graph @ [Ws^T|Wt^T] + bias ----------------
__global__ __launch_bounds__(256) void k_st(const float* __restrict__ Ws_b,
                                            const float* __restrict__ Wt_b,
                                            float* __restrict__ ws) {
  __shared__ float As[16][HH];
  int b = blockIdx.x / MT, m0 = (blockIdx.x % MT) * 16;
  const float* hg = ws + OFF_HGRAPH;
  const float* Wst = ws + OFF_WST;
  float* st = ws + OFF_ST;
  for (int t = threadIdx.x; t < 16 * HH; t += 256) {
    int r = t >> 7, kk = t & 127;
    As[r][kk] = hg[(b * NPAD + m0 + r) * HH + kk];
  }
  __syncthreads();
  int lane = threadIdx.x & 31, wave = threadIdx.x >> 5;
  int mn = lane & 15, kh = (lane >> 4) << 1;
  int col0 = wave * 32;                      // 2 tiles per wave -> 256 cols
  v8f acc[2] = {};
  for (int k = 0; k < HH; k += 4) {
    v2f a; a.x = As[mn][k + kh]; a.y = As[mn][k + kh + 1];
#pragma unroll
    for (int t2 = 0; t2 < 2; ++t2) {
      int cc = col0 + t2 * 16 + mn;
      v2f bf; bf.x = Wst[(k + kh) * 256 + cc]; bf.y = Wst[(k + kh + 1) * 256 + cc];
      acc[t2] = wmma4(a, bf, acc[t2]);
    }
  }
#pragma unroll
  for (int t2 = 0; t2 < 2; ++t2) {
    int cc = col0 + t2 * 16 + mn;
    float bias = (cc < HH) ? Ws_b[cc] : Wt_b[cc - HH];
#pragma unroll
    for (int r = 0; r < 8; ++r)
      st[(b * NPAD + m0 + (lane >> 4) * 8 + r) * 256 + cc] = acc[t2][r] + bias;
  }
}

// ---------------- K4: sparse attention + softmax + context + LayerNorm ----------------
__global__ __launch_bounds__(256) void k_attn(const float* __restrict__ adj,
                                              const float* __restrict__ vvec,
                                              const float* __restrict__ ln_g,
                                              const float* __restrict__ ln_b,
                                              float* __restrict__ ws) {
  int b = blockIdx.x / NN, i = blockIdx.x % NN;
  const float* st = ws + OFF_ST;
  const float* hg = ws + OFF_HGRAPH;
  float* hatt = ws + OFF_HATT;
  __shared__ float s_i[HH];
  __shared__ int jlist[NN];
  __shared__ float sc[NN];
  __shared__ float red[8];
  __shared__ float ctxs[256];
  __shared__ float cf[HH];
  __shared__ int cnt;
  int tid = threadIdx.x, lane = tid & 31, wave = tid >> 5;
  if (tid == 0) cnt = 0;
  if (tid < HH) s_i[tid] = st[(size_t)(b * NPAD + i) * 256 + tid];
  __syncthreads();
  // compact neighbor list (adj != 0)
  for (int j = tid; j < NN; j += 256)
    if (adj[(b * NN + i) * NN + j] != 0.0f) jlist[atomicAdd(&cnt, 1)] = j;
  __syncthreads();
  int M = cnt;
  // scores: one wave per neighbor; lane handles 4 h-dims; shuffle reduce
  for (int p = wave; p < M; p += 8) {
    const float* tj = st + (size_t)(b * NPAD + jlist[p]) * 256 + HH;
    float a = 0.f;
#pragma unroll
    for (int q = 0; q < 4; ++q) {
      int hh = lane * 4 + q;
      a += vvec[hh] * tanhf(s_i[hh] + tj[hh]);
    }
#pragma unroll
    for (int off = 16; off >= 1; off >>= 1) a += __shfl_xor(a, off, 32);
    if (lane == 0) sc[p] = a;
  }
  __syncthreads();
  // softmax over compacted scores (masked entries underflow to 0 exactly)
  float mx = -3.0e38f;
  for (int p = tid; p < M; p += 256) mx = fmaxf(mx, sc[p]);
#pragma unroll
  for (int off = 16; off >= 1; off >>= 1) mx = fmaxf(mx, __shfl_xor(mx, off, 32));
  if (lane == 0) red[wave] = mx;
  __syncthreads();
  if (tid == 0) {
    float m = red[0];
#pragma unroll
    for (int w = 1; w < 8; ++w) m = fmaxf(m, red[w]);
    red[0] = m;
  }
  __syncthreads();
  mx = red[0];
  float sm = 0.f;
  for (int p = tid; p < M; p += 256) { float e = expf(sc[p] - mx); sc[p] = e; sm += e; }
#pragma unroll
  for (int off = 16; off >= 1; off >>= 1) sm += __shfl_xor(sm, off, 32);
  if (lane == 0) red[wave] = sm;
  __syncthreads();
  if (tid == 0) {
    float s2 = 0.f;
#pragma unroll
    for (int w = 0; w < 8; ++w) s2 += red[w];
    red[0] = 1.0f / s2;
  }
  __syncthreads();
  float inv = red[0];
  // context: two half-lists, 128 h-dims each
  int hh = tid & 127, half = tid >> 7;
  float ctx = 0.f;
  for (int p = half; p < M; p += 2)
    ctx += sc[p] * hg[(size_t)(b * NPAD + jlist[p]) * HH + hh];
  ctxs[tid] = ctx;
  __syncthreads();
  if (tid < HH) cf[tid] = (ctxs[tid] + ctxs[tid + HH]) * inv;
  __syncthreads();
  // LayerNorm stats via wave 0
  if (wave == 0) {
    float s = 0.f, sq = 0.f;
#pragma unroll
    for (int q = 0; q < 4; ++q) { float vv = cf[lane * 4 + q]; s += vv; sq += vv * vv; }
#pragma unroll
    for (int off = 16; off >= 1; off >>= 1) { s += __shfl_xor(s, off, 32); sq += __shfl_xor(sq, off, 32); }
    if (lane == 0) { float mu = s * (1.0f / HH); red[0] = mu; red[1] = sq * (1.0f / HH) - mu * mu; }
  }
  __syncthreads();
  float mu = red[0];
  float rstd = rsqrtf(red[1] + 1e-5f);
  if (tid < HH)
    hatt[(size_t)(b * NPAD + i) * HH + tid] = ln_g[tid] * (cf[tid] - mu) * rstd + ln_b[tid];
}

// ---------------- K5: h_new = [h_lstm|h_att] @ comb_w^T + comb_b ----------------
__global__ __launch_bounds__(256) void k_comb(const float* __restrict__ comb_b,
                                              float* __restrict__ ws,
                                              float* __restrict__ out) {
  __shared__ float As[16][256];   // 16 KB
  int b = blockIdx.x / MT, m0 = (blockIdx.x % MT) * 16;
  const float* hlstm = ws + OFF_HLSTM;
  const float* hatt = ws + OFF_HATT;
  const float* Wc = ws + OFF_WCOMB;
  for (int t = threadIdx.x; t < 16 * 256; t += 256) {
    int r = t >> 8, kk = t & 255;
    int row = m0 + r;
    float vv = 0.f;
    if (row < NN) vv = (kk < HH) ? hlstm[(b * NPAD + row) * HH + kk]
                                 : hatt[(b * NPAD + row) * HH + (kk - HH)];
    As[r][kk] = vv;
  }
  __syncthreads();
  int lane = threadIdx.x & 31, wave = threadIdx.x >> 5;
  int mn = lane & 15, kh = (lane >> 4) << 1;
  int col = wave * 16 + mn;
  v8f acc = {};
  for (int k = 0; k < 256; k += 4) {
    v2f a; a.x = As[mn][k + kh]; a.y = As[mn][k + kh + 1];
    v2f bf; bf.x = Wc[(k + kh) * HH + col]; bf.y = Wc[(k + kh + 1) * HH + col];
    acc = wmma4(a, bf, acc);
  }
  float bias = comb_b[col];
#pragma unroll
  for (int r = 0; r < 8; ++r) {
    int row = m0 + (lane >> 4) * 8 + r;
    if (row < NN) out[(b * NN + row) * HH + col] = acc[r] + bias;
  }
}

extern "C" void kernel_launch(void* const* d_in, const int* in_sizes, int n_in,
                              void* d_out, int out_size, void* d_ws, size_t ws_size,
                              hipStream_t stream) {
  (void)in_sizes; (void)n_in; (void)out_size; (void)ws_size;
  const float* x      = (const float*)d_in[0];
  const float* adj    = (const float*)d_in[1];
  const float* h      = (const float*)d_in[2];
  const float* c      = (const float*)d_in[3];
  const float* W_ih   = (const float*)d_in[4];
  const float* W_hh   = (const float*)d_in[5];
  const float* b_ih   = (const float*)d_in[6];
  const float* b_hh   = (const float*)d_in[7];
  const float* gc_w   = (const float*)d_in[8];
  const float* gc_b   = (const float*)d_in[9];
  const float* Ws_w   = (const float*)d_in[10];
  const float* Ws_b   = (const float*)d_in[11];
  const float* Wt_w   = (const float*)d_in[12];
  const float* Wt_b   = (const float*)d_in[13];
  const float* vvec   = (const float*)d_in[14];
  const float* ln_g   = (const float*)d_in[15];
  const float* ln_b   = (const float*)d_in[16];
  const float* comb_w = (const float*)d_in[17];
  const float* comb_b = (const float*)d_in[18];
  float* out = (float*)d_out;
  float* ws = (float*)d_ws;
  float* c_out = out + (size_t)BB * NN * HH;

  k_pack<<<(KL * G4 + 255) / 256, 256, 0, stream>>>(W_ih, W_hh, Ws_w, Wt_w, comb_w, ws);
  k_lstm<<<BB * MT, 256, 0, stream>>>(x, h, c, b_ih, b_hh, ws, c_out);
  k_support<<<BB * MT, 256, 0, stream>>>(gc_w, ws);
  k_graph<<<BB * MT, 256, 0, stream>>>(adj, gc_b, ws);
  k_st<<<BB * MT, 256, 0, stream>>>(Ws_b, Wt_b, ws);
  k_attn<<<BB * NN, 256, 0, stream>>>(adj, vvec, ln_g, ln_b, ws);
  k_comb<<<BB * MT, 256, 0, stream>>>(comb_b, ws, out);
}